// CornerHeadWraper_1202590843775
// MI455X (gfx1250) — compile-verified
//
#include <hip/hip_runtime.h>
#include <hip/hip_bf16.h>

#define BATCH 16
#define NCLS  80
#define HH    128
#define WW    128
#define HWSZ  (HH*WW)           // 16384
#define CHW   (NCLS*HWSZ)       // 1310720
#define KTOP  100
#define NBINS 2048
#define CAND_CAP 4096
#define NDETS 1000
#define SORT2 16384
#define SCORE_THR_C 0.05f
#define IOU_THR_C   0.5f
#define DIST_THR_C  0.5f

typedef __attribute__((ext_vector_type(2))) float v2f;
typedef __attribute__((ext_vector_type(8))) float v8f;

// ---------- key packing: order by score desc, tie -> smaller index first ----------
__device__ __forceinline__ unsigned long long pack_key(float s, unsigned idx) {
    unsigned u = __float_as_uint(s);
    unsigned k = (u & 0x80000000u) ? ~u : (u | 0x80000000u);   // monotonic for all floats
    return (((unsigned long long)k) << 32) | (unsigned long long)(0xFFFFFFFFu - idx);
}
__device__ __forceinline__ float unpack_score(unsigned long long key) {
    unsigned k = (unsigned)(key >> 32);
    unsigned u = (k & 0x80000000u) ? (k ^ 0x80000000u) : ~k;
    return __uint_as_float(u);
}
__device__ __forceinline__ unsigned unpack_idx(unsigned long long key) {
    return 0xFFFFFFFFu - (unsigned)(key & 0xFFFFFFFFu);
}

// ---------- in-LDS bitonic sort, descending, N = power of two ----------
__device__ __forceinline__ void bitonic_desc(unsigned long long* k, int N) {
    const int tid = threadIdx.x, nthr = blockDim.x;
    for (int size = 2; size <= N; size <<= 1) {
        for (int stride = size >> 1; stride > 0; stride >>= 1) {
            __syncthreads();
            for (int i = tid; i < (N >> 1); i += nthr) {
                int a = ((i / stride) * (stride << 1)) + (i % stride);
                int b = a + stride;
                bool desc = ((a & size) == 0);
                unsigned long long ka = k[a], kb = k[b];
                if (desc ? (ka < kb) : (ka > kb)) { k[a] = kb; k[b] = ka; }
            }
        }
    }
    __syncthreads();
}

// sigmoid is monotonic -> local-max test on raw logits is equivalent
__device__ __forceinline__ bool is_peak(const float* plane, int y, int x, float raw) {
    for (int dy = -1; dy <= 1; ++dy) {
        int yy = y + dy; if (yy < 0 || yy >= HH) continue;
        for (int dx = -1; dx <= 1; ++dx) {
            int xx = x + dx;
            if ((dy == 0 && dx == 0) || xx < 0 || xx >= WW) continue;
            if (plane[yy * WW + xx] > raw) return false;
        }
    }
    return true;
}

// ---------- pass 1: peak detect + score histogram (per batch*side) ----------
__global__ void k_hist(const float* __restrict__ tl, const float* __restrict__ br,
                       unsigned* __restrict__ hist) {
    __shared__ unsigned lh[NBINS];
    const int tid = threadIdx.x;
    for (int i = tid; i < NBINS; i += blockDim.x) lh[i] = 0u;
    __syncthreads();
    const int bs = blockIdx.y, b = bs >> 1, side = bs & 1;
    const float* base = (side ? br : tl) + (size_t)b * CHW;
    const int chunk = blockIdx.x;
    for (int i = 0; i < 4; ++i) {
        int e = chunk * 1024 + i * 256 + tid;               // < 1310720 exactly
        int c = e >> 14, rem = e & (HWSZ - 1);
        int y = rem >> 7, x = rem & (WW - 1);
        const float* plane = base + (size_t)c * HWSZ;
        float raw = plane[rem];
        __builtin_prefetch(plane + rem + WW, 0, 1);         // global_prefetch_b8
        if (is_peak(plane, y, x, raw)) {
            float sig = 1.0f / (1.0f + expf(-raw));
            int bin = (int)(sig * (float)NBINS);
            bin = bin < 0 ? 0 : (bin > NBINS - 1 ? NBINS - 1 : bin);
            atomicAdd(&lh[bin], 1u);
        }
    }
    __syncthreads();
    for (int i = tid; i < NBINS; i += blockDim.x)
        if (lh[i]) atomicAdd(&hist[bs * NBINS + i], lh[i]);
}

// ---------- threshold bin so that suffix count >= KTOP ----------
__global__ void k_thresh(const unsigned* __restrict__ hist, int* __restrict__ thr) {
    if (threadIdx.x == 0) {
        const unsigned* h = hist + blockIdx.x * NBINS;
        unsigned acc = 0; int t = 0;
        for (int i = NBINS - 1; i >= 0; --i) {
            acc += h[i];
            if (acc >= (unsigned)KTOP) { t = i; break; }
        }
        thr[blockIdx.x] = t;
    }
}

// ---------- pass 2: collect candidates above threshold ----------
__global__ void k_collect(const float* __restrict__ tl, const float* __restrict__ br,
                          const int* __restrict__ thr, unsigned* __restrict__ cnt,
                          unsigned long long* __restrict__ cand) {
    const int tid = threadIdx.x;
    const int bs = blockIdx.y, b = bs >> 1, side = bs & 1;
    const float* base = (side ? br : tl) + (size_t)b * CHW;
    const int tb = thr[bs];
    const int chunk = blockIdx.x;
    for (int i = 0; i < 4; ++i) {
        int e = chunk * 1024 + i * 256 + tid;
        int c = e >> 14, rem = e & (HWSZ - 1);
        int y = rem >> 7, x = rem & (WW - 1);
        const float* plane = base + (size_t)c * HWSZ;
        float raw = plane[rem];
        __builtin_prefetch(plane + rem + WW, 0, 1);
        if (is_peak(plane, y, x, raw)) {
            float sig = 1.0f / (1.0f + expf(-raw));
            int bin = (int)(sig * (float)NBINS);
            bin = bin < 0 ? 0 : (bin > NBINS - 1 ? NBINS - 1 : bin);
            if (bin >= tb) {
                unsigned pos = atomicAdd(&cnt[bs], 1u);
                if (pos < CAND_CAP)
                    cand[(size_t)bs * CAND_CAP + pos] = pack_key(sig, (unsigned)e);
            }
        }
    }
}

// ---------- exact top-100 per (batch, side): bitonic sort 4096 in LDS ----------
__global__ void k_top100(const unsigned long long* __restrict__ cand,
                         const unsigned* __restrict__ cnt,
                         float* __restrict__ tsc, int* __restrict__ tix) {
    __shared__ unsigned long long sk[CAND_CAP];
    const int tid = threadIdx.x, bs = blockIdx.x;
    unsigned n = cnt[bs]; if (n > CAND_CAP) n = CAND_CAP;
    for (int i = tid; i < CAND_CAP; i += blockDim.x)
        sk[i] = (i < (int)n) ? cand[(size_t)bs * CAND_CAP + i] : 0ULL;
    bitonic_desc(sk, CAND_CAP);
    if (tid < KTOP) {
        unsigned long long key = sk[tid];
        float s = 0.f; unsigned idx = 0u;
        if (key != 0ULL) { s = unpack_score(key); idx = unpack_idx(key); }
        tsc[bs * KTOP + tid] = s;
        tix[bs * KTOP + tid] = (int)idx;
    }
}

// ---------- decode corner positions: gather offsets + embeddings ----------
__global__ void k_decode(const int* __restrict__ tix,
                         const float* __restrict__ tloff, const float* __restrict__ broff,
                         const float* __restrict__ tlemb, const float* __restrict__ bremb,
                         const int* __restrict__ inph, const int* __restrict__ inpw,
                         float* __restrict__ cxs, float* __restrict__ cys,
                         float* __restrict__ cemb, int* __restrict__ ccls) {
    const int tid = threadIdx.x, bs = blockIdx.x, b = bs >> 1, side = bs & 1;
    const float* off = side ? broff : tloff;
    const float* emb = side ? bremb : tlemb;
    const float wr = (float)(*inpw) / (float)WW;
    const float hr = (float)(*inph) / (float)HH;
    if (tid < KTOP) {
        int idx = tix[bs * KTOP + tid];
        int rem = idx & (HWSZ - 1);
        int y = rem >> 7, x = rem & (WW - 1);
        float o0 = off[((size_t)(b * 2 + 0)) * HWSZ + rem];
        float o1 = off[((size_t)(b * 2 + 1)) * HWSZ + rem];
        cxs[bs * KTOP + tid]  = fmaxf(0.f, ((float)x + o0) * wr);
        cys[bs * KTOP + tid]  = fmaxf(0.f, ((float)y + o1) * hr);
        cemb[bs * KTOP + tid] = emb[(size_t)b * HWSZ + rem];
        ccls[bs * KTOP + tid] = idx >> 14;                   // idx / (H*W)
    }
}

// ---------- 100x100 pair scores via V_WMMA_F32_16X16X4_F32 + validity mask ----------
// D[m][n] = 0.5*tl_s[m] + 0.5*br_s[n]  ==  [0.5*tl_s, 0.5] (16x4) x [[1],[br_s]] (4x16)
__global__ void k_score(const float* __restrict__ tsc,
                        const float* __restrict__ cxs, const float* __restrict__ cys,
                        const float* __restrict__ cemb, const int* __restrict__ ccls,
                        float* __restrict__ S) {
    const int tid = threadIdx.x, wave = tid >> 5, lane = tid & 31;
    const int b = blockIdx.y;
    const int tile = blockIdx.x * 8 + wave;                  // 49 real tiles (7x7 of 16)
    const int t = (tile < 49) ? tile : 0;
    const int tm = t / 7, tn = t % 7;
    const float* ts0 = tsc + (b * 2 + 0) * KTOP;
    const float* ts1 = tsc + (b * 2 + 1) * KTOP;
    const int mr = lane & 15;
    const int m0 = tm * 16 + mr;
    const int n0 = tn * 16 + mr;

    v2f a, bb;
    if (lane < 16) {                                         // A rows K=0,1 ; B rows K=0,1
        a[0]  = (m0 < KTOP) ? 0.5f * ts0[m0] : 0.f;
        a[1]  = 0.5f;
        bb[0] = 1.0f;
        bb[1] = (n0 < KTOP) ? ts1[n0] : 0.f;
    } else {                                                 // K=2,3 rows are zero
        a[0] = 0.f; a[1] = 0.f; bb[0] = 0.f; bb[1] = 0.f;
    }
    v8f d = {0.f, 0.f, 0.f, 0.f, 0.f, 0.f, 0.f, 0.f};
    d = __builtin_amdgcn_wmma_f32_16x16x4_f32(false, a, false, bb, (short)0, d,
                                              false, false);
    if (tile < 49) {
        const float* x0 = cxs  + (b * 2 + 0) * KTOP;
        const float* y0 = cys  + (b * 2 + 0) * KTOP;
        const float* e0 = cemb + (b * 2 + 0) * KTOP;
        const int*   c0 = ccls + (b * 2 + 0) * KTOP;
        const float* x1 = cxs  + (b * 2 + 1) * KTOP;
        const float* y1 = cys  + (b * 2 + 1) * KTOP;
        const float* e1 = cemb + (b * 2 + 1) * KTOP;
        const int*   c1 = ccls + (b * 2 + 1) * KTOP;
        for (int r = 0; r < 8; ++r) {
            int m = tm * 16 + r + ((lane < 16) ? 0 : 8);     // C/D VGPR layout
            int n = tn * 16 + mr;
            if (m < KTOP && n < KTOP) {
                bool neg = (c0[m] != c1[n]) || (x1[n] <= x0[m]) || (y1[n] <= y0[m]) ||
                           (fabsf(e0[m] - e1[n]) > DIST_THR_C);
                S[(size_t)b * (KTOP * KTOP) + m * KTOP + n] = neg ? -1.0f : d[r];
            }
        }
    }
}

// ---------- exact sorted top-1000: bitonic sort 16384 keys in 128KB LDS ----------
__global__ void k_top1000(const float* __restrict__ S,
                          const float* __restrict__ cxs, const float* __restrict__ cys,
                          const int* __restrict__ ccls,
                          float* __restrict__ dx1, float* __restrict__ dy1,
                          float* __restrict__ dx2, float* __restrict__ dy2,
                          float* __restrict__ dsc, int* __restrict__ dcl) {
    extern __shared__ unsigned long long dsk[];
    const int tid = threadIdx.x, b = blockIdx.x;
    for (int i = tid; i < SORT2; i += blockDim.x) {
        unsigned long long key = 0ULL;
        if (i < KTOP * KTOP) key = pack_key(S[(size_t)b * (KTOP * KTOP) + i], (unsigned)i);
        dsk[i] = key;
    }
    bitonic_desc(dsk, SORT2);
    for (int j = tid; j < NDETS; j += blockDim.x) {
        unsigned long long key = dsk[j];
        float s = unpack_score(key);
        unsigned idx = unpack_idx(key);
        int m = (int)(idx / KTOP), n = (int)(idx % KTOP);
        dx1[b * NDETS + j] = cxs[(b * 2 + 0) * KTOP + m];
        dy1[b * NDETS + j] = cys[(b * 2 + 0) * KTOP + m];
        dx2[b * NDETS + j] = cxs[(b * 2 + 1) * KTOP + n];
        dy2[b * NDETS + j] = cys[(b * 2 + 1) * KTOP + n];
        dsc[b * NDETS + j] = s;
        dcl[b * NDETS + j] = ccls[(b * 2 + 0) * KTOP + m];
    }
}

// ---------- greedy class-aware NMS + final top-100 emission ----------
__global__ void k_nms(const float* __restrict__ dx1, const float* __restrict__ dy1,
                      const float* __restrict__ dx2, const float* __restrict__ dy2,
                      const float* __restrict__ dsc, const int* __restrict__ dcl,
                      float* __restrict__ out) {
    __shared__ float sx1[1024], sy1[1024], sx2[1024], sy2[1024], ssc[1024];
    __shared__ int scl[1024], skp[1024];
    const int tid = threadIdx.x, b = blockIdx.x;
    if (tid < NDETS) {
        sx1[tid] = dx1[b * NDETS + tid]; sy1[tid] = dy1[b * NDETS + tid];
        sx2[tid] = dx2[b * NDETS + tid]; sy2[tid] = dy2[b * NDETS + tid];
        ssc[tid] = dsc[b * NDETS + tid]; scl[tid] = dcl[b * NDETS + tid];
        skp[tid] = 1;
    } else { ssc[tid] = -1.f; skp[tid] = 0; scl[tid] = -2; sx1[tid]=sy1[tid]=sx2[tid]=sy2[tid]=0.f; }
    __syncthreads();
    for (int i = 0; i < NDETS; ++i) {
        bool ki = (skp[i] != 0) && (ssc[i] > SCORE_THR_C);
        if (ki && tid > i && tid < NDETS && skp[tid] && scl[tid] == scl[i]) {
            float ai = fmaxf(sx2[i]  - sx1[i],  0.f) * fmaxf(sy2[i]  - sy1[i],  0.f);
            float aj = fmaxf(sx2[tid] - sx1[tid], 0.f) * fmaxf(sy2[tid] - sy1[tid], 0.f);
            float iw = fmaxf(fminf(sx2[i], sx2[tid]) - fmaxf(sx1[i], sx1[tid]), 0.f);
            float ih = fmaxf(fminf(sy2[i], sy2[tid]) - fmaxf(sy1[i], sy1[tid]), 0.f);
            float inter = iw * ih;
            float iou = inter / fmaxf(ai + aj - inter, 1e-6f);
            if (iou > IOU_THR_C) skp[tid] = 0;
        }
        __syncthreads();
    }
    if (tid == 0) {
        float* ob = out + BATCH + b * (KTOP * 4);                       // boxes
        float* os = out + BATCH + BATCH * KTOP * 4 + b * KTOP;          // scores
        float* oc = out + BATCH + BATCH * KTOP * 5 + b * KTOP;          // classes
        int total = 0, w = 0;
        for (int j = 0; j < NDETS; ++j) {
            if (skp[j] && ssc[j] > SCORE_THR_C) {
                if (w < KTOP) {
                    ob[4 * w + 0] = sx1[j]; ob[4 * w + 1] = sy1[j];
                    ob[4 * w + 2] = sx2[j]; ob[4 * w + 3] = sy2[j];
                    os[w] = ssc[j]; oc[w] = (float)scl[j]; ++w;
                }
                ++total;
            }
        }
        for (; w < KTOP; ++w) {
            ob[4 * w + 0] = ob[4 * w + 1] = ob[4 * w + 2] = ob[4 * w + 3] = 0.f;
            os[w] = 0.f; oc[w] = -1.f;
        }
        out[b] = (float)(total < KTOP ? total : KTOP);
    }
}

extern "C" void kernel_launch(void* const* d_in, const int* in_sizes, int n_in,
                              void* d_out, int out_size, void* d_ws, size_t ws_size,
                              hipStream_t stream) {
    (void)in_sizes; (void)n_in; (void)out_size; (void)ws_size;
    const float* tl    = (const float*)d_in[0];
    const float* br    = (const float*)d_in[1];
    const float* tloff = (const float*)d_in[2];
    const float* broff = (const float*)d_in[3];
    const float* tlemb = (const float*)d_in[4];
    const float* bremb = (const float*)d_in[5];
    const int*   inph  = (const int*)d_in[6];
    const int*   inpw  = (const int*)d_in[7];

    char* ws = (char*)d_ws;
    size_t o = 0;
    auto alloc = [&](size_t bytes) { size_t r = o; o = (o + bytes + 255) & ~(size_t)255; return r; };

    unsigned*            hist = (unsigned*)(ws + alloc(32 * NBINS * sizeof(unsigned)));
    int*                 thr  = (int*)(ws + alloc(32 * sizeof(int)));
    unsigned*            cnt  = (unsigned*)(ws + alloc(32 * sizeof(unsigned)));
    const size_t zbytes = o;                                  // hist+thr+cnt zeroed each call
    unsigned long long*  cand = (unsigned long long*)(ws + alloc((size_t)32 * CAND_CAP * 8));
    float*               tsc  = (float*)(ws + alloc(32 * KTOP * sizeof(float)));
    int*                 tix  = (int*)(ws + alloc(32 * KTOP * sizeof(int)));
    float*               cxs  = (float*)(ws + alloc(32 * KTOP * sizeof(float)));
    float*               cys  = (float*)(ws + alloc(32 * KTOP * sizeof(float)));
    float*               cemb = (float*)(ws + alloc(32 * KTOP * sizeof(float)));
    int*                 ccls = (int*)(ws + alloc(32 * KTOP * sizeof(int)));
    float*               S    = (float*)(ws + alloc((size_t)BATCH * KTOP * KTOP * sizeof(float)));
    float*               dx1  = (float*)(ws + alloc(BATCH * NDETS * sizeof(float)));
    float*               dy1  = (float*)(ws + alloc(BATCH * NDETS * sizeof(float)));
    float*               dx2  = (float*)(ws + alloc(BATCH * NDETS * sizeof(float)));
    float*               dy2  = (float*)(ws + alloc(BATCH * NDETS * sizeof(float)));
    float*               dsc  = (float*)(ws + alloc(BATCH * NDETS * sizeof(float)));
    int*                 dcl  = (int*)(ws + alloc(BATCH * NDETS * sizeof(int)));

    (void)hipMemsetAsync(ws, 0, zbytes, stream);
    (void)hipFuncSetAttribute((const void*)k_top1000,
                              hipFuncAttributeMaxDynamicSharedMemorySize, SORT2 * 8);

    k_hist   <<<dim3(1280, 32), 256, 0, stream>>>(tl, br, hist);
    k_thresh <<<32, 64, 0, stream>>>(hist, thr);
    k_collect<<<dim3(1280, 32), 256, 0, stream>>>(tl, br, thr, cnt, cand);
    k_top100 <<<32, 1024, 0, stream>>>(cand, cnt, tsc, tix);
    k_decode <<<32, 128, 0, stream>>>(tix, tloff, broff, tlemb, bremb, inph, inpw,
                                      cxs, cys, cemb, ccls);
    k_score  <<<dim3(7, BATCH), 256, 0, stream>>>(tsc, cxs, cys, cemb, ccls, S);
    k_top1000<<<BATCH, 1024, SORT2 * 8, stream>>>(S, cxs, cys, ccls,
                                                  dx1, dy1, dx2, dy2, dsc, dcl);
    k_nms    <<<BATCH, 1024, 0, stream>>>(dx1, dy1, dx2, dy2, dsc, dcl, (float*)d_out);
}